// ExcitationGeneration_64836826301036
// MI455X (gfx1250) — compile-verified
//
#include <hip/hip_runtime.h>
#include <math.h>

#define FRAMES 4000
#define FP 120
#define NS (FRAMES * FP)
#define BASE_STRIDE 4096
#define EPSF 1e-8f

typedef float v2f __attribute__((ext_vector_type(2)));
typedef float v8f __attribute__((ext_vector_type(8)));

#define HAVE_WMMA_F32X4 __has_builtin(__builtin_amdgcn_wmma_f32_16x16x4_f32)

// ---------------- fast math helpers (straight-line, no exec-mask branching) ---
__device__ __forceinline__ float fast_rcp(float x) {
#if __has_builtin(__builtin_amdgcn_rcpf)
  float r = __builtin_amdgcn_rcpf(x);       // v_rcp_f32 (~1 ulp)
  r = r * (2.0f - x * r);                   // one Newton step -> ~0.5 ulp
  return r;
#else
  return 1.0f / x;
#endif
}

__device__ __forceinline__ float fast_sqrt(float x) {
#if __has_builtin(__builtin_amdgcn_sqrtf)
  return __builtin_amdgcn_sqrtf(x);         // v_sqrt_f32
#else
  return sqrtf(x);
#endif
}

__device__ __forceinline__ float fast_ln(float x) {
#if __has_builtin(__builtin_amdgcn_logf)
  return __builtin_amdgcn_logf(x) * 0.6931471805599453f; // v_log_f32 is log2
#else
  return logf(x);
#endif
}

__device__ __forceinline__ float fast_cos2pi(float u) {  // cos(2*pi*u), u in [0,1)
#if __has_builtin(__builtin_amdgcn_cosf)
  return __builtin_amdgcn_cosf(u);          // v_cos_f32 input is revolutions
#else
  return cosf(6.28318530717958647f * u);
#endif
}

// ---------------- noise (deterministic hash -> Box-Muller) ----------------
__device__ __forceinline__ unsigned hashu(unsigned x) {
  x ^= x >> 16; x *= 0x7feb352dU;
  x ^= x >> 15; x *= 0x846ca68bU;
  x ^= x >> 16;
  return x;
}

__device__ __forceinline__ float gauss_noise(unsigned idx) {
  unsigned h1 = hashu(idx * 2u + 0x9E3779B9u);
  unsigned h2 = hashu(idx * 2u + 0x85EBCA6Bu);
  float u1 = ((float)(h1 >> 8) + 0.5f) * (1.0f / 16777216.0f); // (0,1)
  float u2 = ((float)(h2 >> 8)) * (1.0f / 16777216.0f);        // [0,1)
  float r = fast_sqrt(-2.0f * fast_ln(u1));
  return r * fast_cos2pi(u2);
}

// ---------------- Phase 1: per-frame Q sums + segmented exclusive scan ----
// One block (256 threads) per row; each thread owns 16 consecutive frames.
__global__ __launch_bounds__(256) void eg_phase1(const float* __restrict__ p,
                                                 float* __restrict__ base) {
  const int row = blockIdx.x;
  const float* pr = p + row * FRAMES;
  float* br = base + row * BASE_STRIDE;
  const int tid = threadIdx.x;
  const int f0 = tid * 16;

  float Q[16];
  unsigned vmask = 0u;
  #pragma unroll
  for (int k = 0; k < 16; ++k) {
    const int i = f0 + k;
    const float pi = (i < FRAMES) ? pr[i] : 0.0f;
    const bool v = (i < FRAMES) && (pi >= 1.0f);
    float Qi = 0.0f;
    if (v) {
      const float pn1 = (i + 1 < FRAMES) ? pr[i + 1] : 0.0f;
      const float pn = (pn1 >= 1.0f) ? pn1 : pi; // voiced->unvoiced: hold pitch
      const float dp = (pn - pi) * (1.0f / (float)FP);
      #pragma unroll 8
      for (int j = 0; j < FP; ++j) Qi += fast_rcp(pi + dp * (float)j);
      vmask |= (1u << k);
    }
    Q[k] = Qi;
  }

  // Local summary: T = sum of Q; M = running-prefix value at last unvoiced
  // frame in this chunk (-inf if none). Operator is associative:
  // (T1,M1)o(T2,M2) = (T1+T2, max(M1, T1+M2)).
  float T = 0.0f, M = -INFINITY;
  #pragma unroll
  for (int k = 0; k < 16; ++k) {
    T += Q[k];
    if (!((vmask >> k) & 1u)) M = fmaxf(M, T);
  }

  __shared__ float sT[256];
  __shared__ float sM[256];
  sT[tid] = T; sM[tid] = M;
  __syncthreads();
  for (int off = 1; off < 256; off <<= 1) {
    float tl = 0.0f, ml = -INFINITY;
    const float Tc = sT[tid];
    const float Mc = sM[tid];
    if (tid >= off) { tl = sT[tid - off]; ml = sM[tid - off]; }
    __syncthreads();
    if (tid >= off) {
      sT[tid] = tl + Tc;
      sM[tid] = (Mc == -INFINITY) ? ml : fmaxf(ml, tl + Mc);
    }
    __syncthreads();
  }

  float S  = (tid > 0) ? sT[tid - 1] : 0.0f;        // exclusive prefix sum
  float Mi = (tid > 0) ? sM[tid - 1] : -INFINITY;
  float Bv = fmaxf(0.0f, Mi);                       // running bias

  #pragma unroll
  for (int k = 0; k < 16; ++k) {
    const int i = f0 + k;
    if (i < FRAMES) {
      br[i] = ((vmask >> k) & 1u) ? fmaxf(S - Bv, 0.0f) : 0.0f;
    }
    S += Q[k];
    if (!((vmask >> k) & 1u)) Bv = fmaxf(Bv, S);
  }
}

// ---------------- Phase 2: sample generation, WMMA-based prefix scan ------
// Block = 128 threads (4 waves). Each wave handles 16 frames: column n of a
// 16x16 tile = frame f0+n, rows = samples within a 16-sample chunk.
// D = L * X (L = lower-triangular ones) computed as 4 chained K=4 WMMAs
// gives the within-chunk inclusive prefix for all 16 frames at once.
__global__ __launch_bounds__(128) void eg_phase2(const float* __restrict__ p,
                                                 const float* __restrict__ base,
                                                 float* __restrict__ out) {
  const int lane = threadIdx.x & 31;
  const int wv   = threadIdx.x >> 5;
  const int row  = blockIdx.y;
  const int f0   = blockIdx.x * 64 + wv * 16;
  const int col  = lane & 15;          // frame column within the tile
  const int half = lane >> 4;          // 0: rows 0-7, 1: rows 8-15 (C layout)
  const int fi   = f0 + col;
  const bool inRange = fi < FRAMES;

  const float pi0 = inRange ? p[row * FRAMES + fi] : 0.0f;
  const bool voiced = inRange && (pi0 >= 1.0f);
  float pn;
  {
    const float pn1 = (fi + 1 < FRAMES) ? p[row * FRAMES + fi + 1] : 0.0f;
    pn = (pn1 >= 1.0f) ? pn1 : pi0;
  }
  const float dp = (pn - pi0) * (1.0f / (float)FP);
  const float bs = voiced ? base[row * BASE_STRIDE + fi] : 0.0f;

  const long long outBase = (long long)row * NS + (long long)fi * FP;
  float carry = 0.0f;

  for (int t = 0; t < 8; ++t) {        // 8 chunks of 16 samples (120 + 8 pad)
    float d[8];
#if HAVE_WMMA_F32X4
    {
      v8f acc = {};
      const int mA = lane & 15;        // A-layout: lanes hold M, halves hold K-pairs
      const int kb = half * 2;
      #pragma unroll
      for (int u = 0; u < 4; ++u) {
        const int k0 = 4 * u + kb;
        v2f a, b;
        a.x = (mA >= k0)     ? 1.0f : 0.0f;   // L[:, k0]
        a.y = (mA >= k0 + 1) ? 1.0f : 0.0f;   // L[:, k0+1]
        const int j0 = 16 * t + k0;           // B(k,n) = q_n(16t + k)
        const float r0 = fast_rcp(pi0 + dp * (float)j0);
        const float r1 = fast_rcp(pi0 + dp * (float)(j0 + 1));
        b.x = (voiced && j0     < FP) ? r0 : 0.0f;   // v_cndmask, no branches
        b.y = (voiced && j0 + 1 < FP) ? r1 : 0.0f;
        acc = __builtin_amdgcn_wmma_f32_16x16x4_f32(
            false, a, false, b, (short)0, acc, false, false);
      }
      #pragma unroll
      for (int r = 0; r < 8; ++r) d[r] = acc[r];
    }
#else
    {
      float run = 0.0f;
      #pragma unroll
      for (int r = 0; r < 8; ++r) {
        const int j = 16 * t + half * 8 + r;
        const float rq = fast_rcp(pi0 + dp * (float)j);
        const float q = (voiced && j < FP) ? rq : 0.0f;
        run += q;
        d[r] = run;
      }
      const float oth = __shfl_xor(d[7], 16, 32);
      if (half) {
        #pragma unroll
        for (int r = 0; r < 8; ++r) d[r] += oth;
      }
    }
#endif
    // Cross-half carry: lower gets D(15,n) (chunk total), upper gets D(7,n).
    const float oth7  = __shfl_xor(d[7], 16, 32);
    const float sb    = bs + carry;
    const float total = half ? d[7] : oth7;
    float cp = ceilf(sb + (half ? oth7 : 0.0f) - EPSF); // ceil at previous sample

    float e[8];
    const int jb = 16 * t + half * 8;
    #pragma unroll
    for (int r = 0; r < 8; ++r) {
      const int j = jb + r;
      const float sv = sb + d[r];
      const float cv = ceilf(sv - EPSF);
      const bool pulse = (cv - cp) >= 1.0f;
      cp = cv;
      const float psamp = pi0 + dp * (float)j;
      float val;
      if (voiced) {
        val = pulse ? fast_sqrt(psamp) : 0.0f;
      } else {
        val = gauss_noise((unsigned)(outBase + j));
      }
      e[r] = val;
    }
    carry += total;

    if (inRange && jb < FP) {          // lanes with jb==120 (t=7 upper) skip
      float4 w0 = make_float4(e[0], e[1], e[2], e[3]);
      float4 w1 = make_float4(e[4], e[5], e[6], e[7]);
      *(float4*)(out + outBase + jb)     = w0;
      *(float4*)(out + outBase + jb + 4) = w1;
    }
  }
}

// ---------------- launch ---------------------------------------------------
extern "C" void kernel_launch(void* const* d_in, const int* in_sizes, int n_in,
                              void* d_out, int out_size, void* d_ws, size_t ws_size,
                              hipStream_t stream) {
  (void)n_in; (void)out_size; (void)ws_size;
  const float* p = (const float*)d_in[0];
  float* out = (float*)d_out;
  float* base = (float*)d_ws;                 // 64 * 4096 floats = 1 MB

  const int rows = in_sizes[0] / FRAMES;      // 64

  eg_phase1<<<dim3(rows), dim3(256), 0, stream>>>(p, base);

  // 63 blocks * 64 frames cover 4000 frames; grid.y = rows
  eg_phase2<<<dim3((FRAMES + 63) / 64, rows), dim3(128), 0, stream>>>(p, base, out);
}